// HMM_54846732370239
// MI455X (gfx1250) — compile-verified
//
#include <hip/hip_runtime.h>
#include <math.h>

#define N_TAGS   512
#define N_TOKENS 50257
#define SEQ_LEN  8192

#define NWG   16          // workgroups (one per WGP), cols per WG = 32
#define TPB   512         // 16 wave32 per WG
#define COLS  32          // columns owned per WG
#define RSEG  16          // i-segments (one per wave)
#define ESEG  32          // i-elements per segment (held in VGPRs)

#define BT_K  128         // backtrack block length
#define BT_NB 64          // number of backtrack blocks

typedef unsigned int  u32x4 __attribute__((ext_vector_type(4)));
typedef int           i32x8 __attribute__((ext_vector_type(8)));
typedef int           i32x4 __attribute__((ext_vector_type(4)));

// ---------------------------------------------------------------------------
// ws layout (bytes):
//   [0)      unsigned cnt          (grid-sync generation counter)
//   [512)    float gv[2][512]      (double-buffered v vector)
//   [8192)   u16 bp[8191][512]     (backpointers, 8.0 MB)
//   [+)      u16 comp[64][512]     (block-composed state maps)
//   [+)      int ent[64]           (block entry states)
// ---------------------------------------------------------------------------

__global__ void init_cnt(unsigned* cnt) { *cnt = 0u; }

__global__ __launch_bounds__(TPB)
void viterbi_scan(const int* __restrict__ tok, const float* __restrict__ A,
                  const float* __restrict__ B, const float* __restrict__ Pi,
                  float* __restrict__ gv,            // [2][512]
                  unsigned short* __restrict__ bp,   // [8191][512]
                  unsigned int* __restrict__ cnt)
{
  __shared__ float stage[256 * COLS];   // 32KB TDM staging for the A slice
  __shared__ float vsh[N_TAGS];         // current v vector
  __shared__ float pval[TPB];           // per-segment partial max
  __shared__ int   pidx[TPB];           // per-segment partial argmax

  const int tid = threadIdx.x;
  const int wg  = blockIdx.x;
  const int jl  = tid & (COLS - 1);     // lane id within wave == local column
  const int rs  = tid >> 5;             // wave id == i-segment
  const int j   = wg * COLS + jl;       // global column

  float areg[ESEG];                     // logA[rs*32 .. rs*32+31][j] in VGPRs

  // ---- Stage this WG's logA slice via the Tensor Data Mover (2 chunks) ----
  for (int c = 0; c < 2; ++c) {
    if (tid == 0) {
      unsigned lds = (unsigned)(size_t)(&stage[0]);   // low 32 bits = LDS offset
      unsigned long long ga =
          (unsigned long long)(size_t)(A + ((size_t)c * 256 * N_TAGS + (size_t)wg * COLS));
      u32x4 g0 = { 1u,                                 // count = 1 valid descriptor
                   lds,                                // lds_addr
                   (unsigned)(ga & 0xFFFFFFFFu),       // global_addr[31:0]
                   (unsigned)((ga >> 32) & 0x01FFFFFFu) | (2u << 30) }; // [56:32] | type=2
      i32x8 g1 = { (int)(2u << 16),                    // data_size = 4B
                   0,                                  // tensor_dim0 = 65536 (lo16=0)
                   1,                                  // tensor_dim0 hi16 = 1
                   1 | (COLS << 16),                   // tensor_dim1 hi16=1 | tile_dim0=32
                   256,                                // tile_dim1 = 256 rows, tile_dim2=0
                   N_TAGS,                             // tensor_dim0_stride = 512
                   0, 0 };
      i32x4 gz4 = { 0, 0, 0, 0 };
      i32x8 gz8 = { 0, 0, 0, 0, 0, 0, 0, 0 };
      __builtin_amdgcn_tensor_load_to_lds(g0, g1, gz4, gz4, gz8, 0);
    }
    if (tid < 32) __builtin_amdgcn_s_wait_tensorcnt(0);
    __syncthreads();
    if ((rs >> 3) == c) {               // this wave's rows live in chunk c
      const int lr0 = (rs & 7) * ESEG;
#pragma unroll
      for (int k = 0; k < ESEG; ++k)
        areg[k] = logf(stage[(lr0 + k) * COLS + jl]);
    }
    __syncthreads();
  }

  // ---- v0 = log(Pi) + emission(0) (computed redundantly by every WG) ----
  {
    int t0k = tok[0];
    float e0 = (t0k < 0) ? -logf((float)N_TAGS)
                         : logf(B[(size_t)tid * N_TOKENS + (size_t)t0k]);
    vsh[tid] = logf(Pi[tid]) + e0;
  }
  __syncthreads();

  const float NEG = -3.0e38f;
  for (int t = 1; t < SEQ_LEN; ++t) {
    // Early-issue emission gather (latency hidden under the max-plus phase).
    float Braw = 1.0f; int tk = 0;
    if (tid < COLS) {
      tk = tok[t];
      if (tk >= 0) Braw = B[(size_t)j * N_TOKENS + (size_t)tk];
    }

    // Phase A: per-segment max-plus with argmax, A in VGPRs, v broadcast from LDS.
    float m = NEG; int mi = 0;
    const int ib = rs * ESEG;
#pragma unroll
    for (int k = 0; k < ESEG; ++k) {
      float s = vsh[ib + k] + areg[k];
      if (s > m) { m = s; mi = ib + k; }   // strict > keeps first (lowest i)
    }
    pval[tid] = m; pidx[tid] = mi;
    __syncthreads();

    // Phase B (wave 0): cross-segment reduce, add emission, publish slice.
    if (tid < COLS) {
      float bm = pval[tid]; int bi = pidx[tid];
#pragma unroll
      for (int r = 1; r < RSEG; ++r) {
        float c = pval[r * COLS + tid];
        if (c > bm) { bm = c; bi = pidx[r * COLS + tid]; }  // ascending i blocks
      }
      float e = (tk < 0) ? -logf((float)N_TAGS) : logf(Braw);
      gv[(t & 1) * N_TAGS + j] = bm + e;
      bp[(size_t)(t - 1) * N_TAGS + j] = (unsigned short)bi;
    }

    // Grid sync: release-atomic generation barrier across the 16 WGPs.
    if (tid == 0) {
      __hip_atomic_fetch_add(cnt, 1u, __ATOMIC_RELEASE, __HIP_MEMORY_SCOPE_AGENT);
      const unsigned tgt = (unsigned)t * NWG;
      while (__hip_atomic_load(cnt, __ATOMIC_RELAXED, __HIP_MEMORY_SCOPE_AGENT) < tgt)
        __builtin_amdgcn_s_sleep(1);
    }
    __syncthreads();
    __builtin_amdgcn_fence(__ATOMIC_ACQUIRE, "agent");
    __builtin_amdgcn_s_cluster_barrier();   // s_barrier_signal/wait -3; NOP if not clustered

    // Pull fresh v into LDS via the CDNA5 async global->LDS path.
    {
      unsigned loff = (unsigned)(size_t)(&vsh[tid]);
      const float* gp = gv + (t & 1) * N_TAGS + tid;
      asm volatile("global_load_async_to_lds_b32 %0, %1, off"
                   :: "v"(loff), "v"(gp) : "memory");
      asm volatile("s_wait_asynccnt 0" ::: "memory");
    }
    __syncthreads();
  }
}

// ---- Backtrack stage 1: compose each 128-row block's state map in parallel ----
__global__ __launch_bounds__(N_TAGS)
void bt_compose(const unsigned short* __restrict__ bp, unsigned short* __restrict__ comp)
{
  const int b = blockIdx.x;              // 0..63
  const int s = threadIdx.x;             // 0..511
  const int lo = b * BT_K;
  int hi = (b + 1) * BT_K; if (hi > SEQ_LEN - 1) hi = SEQ_LEN - 1; hi -= 1;
  int p = s;
  for (int t = hi; t >= lo; --t)
    p = bp[(size_t)t * N_TAGS + p];
  comp[b * N_TAGS + s] = (unsigned short)p;
}

// ---- Backtrack stage 2: final argmax + 64-long composed chain ----
__global__ void bt_seed(const float* __restrict__ gv1,
                        const unsigned short* __restrict__ comp,
                        int* __restrict__ ent, int* __restrict__ out)
{
  float bm = -3.0e38f; int bi = 0;
  for (int jj = 0; jj < N_TAGS; ++jj) {
    float v = gv1[jj];
    if (v > bm) { bm = v; bi = jj; }     // first-index tie rule
  }
  out[SEQ_LEN - 1] = bi;
  int e = bi;
  ent[BT_NB - 1] = e;                    // ent[b] = path[(b+1)*K]
  for (int b = BT_NB - 1; b >= 1; --b) {
    e = comp[b * N_TAGS + e];
    ent[b - 1] = e;
  }
}

// ---- Backtrack stage 3: fill each block's path segment in parallel ----
__global__ void bt_fill(const unsigned short* __restrict__ bp,
                        const int* __restrict__ ent, int* __restrict__ out)
{
  const int b = blockIdx.x * blockDim.x + threadIdx.x;
  if (b >= BT_NB) return;
  const int lo = b * BT_K;
  int hi = (b + 1) * BT_K; if (hi > SEQ_LEN - 1) hi = SEQ_LEN - 1; hi -= 1;
  int p = ent[b];
  for (int t = hi; t >= lo; --t) {
    p = bp[(size_t)t * N_TAGS + p];
    out[t] = p;
  }
}

extern "C" void kernel_launch(void* const* d_in, const int* in_sizes, int n_in,
                              void* d_out, int out_size, void* d_ws, size_t ws_size,
                              hipStream_t stream)
{
  const int*   tok = (const int*)d_in[0];    // token_list  [8192] int32
  const float* A   = (const float*)d_in[1];  // A  [512,512]
  const float* B   = (const float*)d_in[2];  // B  [512,50257]
  const float* Pi  = (const float*)d_in[3];  // Pi [512]

  char* ws = (char*)d_ws;
  unsigned* cnt        = (unsigned*)(ws + 0);
  float*    gv         = (float*)(ws + 512);
  unsigned short* bp   = (unsigned short*)(ws + 8192);
  const size_t bpBytes = (size_t)(SEQ_LEN - 1) * N_TAGS * 2;   // 8,387,584
  unsigned short* comp = (unsigned short*)(ws + 8192 + bpBytes);
  int*      ent        = (int*)(ws + 8192 + bpBytes + (size_t)BT_NB * N_TAGS * 2);
  int*      out        = (int*)d_out;

  init_cnt<<<1, 1, 0, stream>>>(cnt);
  viterbi_scan<<<NWG, TPB, 0, stream>>>(tok, A, B, Pi, gv, bp, cnt);
  bt_compose<<<BT_NB, N_TAGS, 0, stream>>>(bp, comp);
  bt_seed<<<1, 1, 0, stream>>>(gv + 1 * N_TAGS, comp, ent, out);  // final buffer = (8191&1)=1
  bt_fill<<<2, 32, 0, stream>>>(bp, ent, out);
}